// MemoryBank3_85770496901141
// MI455X (gfx1250) — compile-verified
//
#include <hip/hip_runtime.h>

#ifndef __has_builtin
#define __has_builtin(x) 0
#endif

#if defined(__gfx1250__) && __has_builtin(__builtin_amdgcn_global_load_async_to_lds_b128) && __has_builtin(__builtin_amdgcn_s_wait_asynccnt)
#define HAVE_ASYNC_LDS 1
#else
#define HAVE_ASYNC_LDS 0
#endif

#define MAXN 128
#define TPB 256

typedef float floatx4 __attribute__((ext_vector_type(4)));   // native 128-bit vector
typedef int   intx4   __attribute__((vector_size(16)));      // for async-LDS builtin

// ---------------------------------------------------------------------------
// Kernel 1: per-step class index: tstep[s] = argmax_j targets[mask[s]][j]
// jnp.argmax semantics: first occurrence of the maximum wins.
// ---------------------------------------------------------------------------
__global__ void mb_argmax_kernel(const float* __restrict__ targets,
                                 const int* __restrict__ mask,
                                 int* __restrict__ tstep,
                                 int C) {
    __shared__ float sv[TPB];
    __shared__ int   si[TPB];
    const int s = blockIdx.x;
    const int m = mask[s];
    const float* row = targets + (long long)m * C;

    float bv = -__builtin_inff();
    int   bi = 0x7fffffff;
    for (int j = threadIdx.x; j < C; j += blockDim.x) {
        float v = row[j];
        if (v > bv) { bv = v; bi = j; }   // ascending j => first occurrence kept
    }
    sv[threadIdx.x] = bv;
    si[threadIdx.x] = bi;
    __syncthreads();
    for (int off = TPB >> 1; off > 0; off >>= 1) {
        if (threadIdx.x < off) {
            float ov = sv[threadIdx.x + off];
            int   oi = si[threadIdx.x + off];
            if (ov > sv[threadIdx.x] ||
                (ov == sv[threadIdx.x] && oi < si[threadIdx.x])) {
                sv[threadIdx.x] = ov;
                si[threadIdx.x] = oi;
            }
        }
        __syncthreads();
    }
    if (threadIdx.x == 0) tstep[s] = si[0];
}

// ---------------------------------------------------------------------------
// Kernel 2: one workgroup per class. Sequentially replays this class's
// accepted updates on (confidence, source-row-id) pairs in LDS, then gathers
// the final 128 x D row block to the output with non-temporal accesses.
//
// Source-row encoding: r < N  -> memory[c][r]
//                      r >= N -> batch_features[r - N]
// ---------------------------------------------------------------------------
__global__ void mb_class_kernel(const float* __restrict__ memory,
                                const float* __restrict__ confidences,
                                const float* __restrict__ feats,
                                const int* __restrict__ bconf,
                                const int* __restrict__ mask,
                                const int* __restrict__ tstep,
                                float* __restrict__ out,
                                int N, int D, int B) {
    __shared__ __align__(16) float cf[MAXN];     // logical confidences (sorted state)
    __shared__ int                 rowid[MAXN];  // logical slot -> physical source row
    __shared__ float               nkey[MAXN];   // scratch: post-shift keys
    __shared__ int                 nrow[MAXN];   // scratch: post-shift row ids

    const int c = blockIdx.x;
    const int t = threadIdx.x;

    // ---- init state: cf = confidences[c], rowid = identity -----------------
#if HAVE_ASYNC_LDS
    if (t < (N >> 2)) {
        __builtin_amdgcn_global_load_async_to_lds_b128(
            (__attribute__((address_space(1))) intx4*)(confidences + (long long)c * N + t * 4),
            (__attribute__((address_space(3))) intx4*)(&cf[t * 4]),
            0, 0);
    }
    if (t < N) rowid[t] = t;
    __builtin_amdgcn_s_wait_asynccnt(0);
#else
    if (t < N) {
        cf[t]    = confidences[(long long)c * N + t];
        rowid[t] = t;
    }
#endif
    __syncthreads();

    // ---- sequential replay of this class's updates -------------------------
    for (int s = 0; s < B; ++s) {
        if (tstep[s] != c) continue;               // uniform (scalar) branch
        const int   m    = mask[s];
        const float conf = (float)bconf[m];
        if (!(conf > cf[N - 1])) continue;         // uniform: shared-mem value

        // Phase 1: shifted rows + overwritten last confidence (NOT shifted),
        // exactly matching the reference's quirk.
        float myk = 0.0f;
        int   myr = 0;
        if (t < N) {
            if (t < N - 1) { myk = cf[t]; myr = rowid[t + 1]; }
            else           { myk = conf; myr = N + m; }
            nkey[t] = myk;
            nrow[t] = myr;
        }
        __syncthreads();

        // Phase 2: stable descending sort via rank computation.
        if (t < N) {
            int rank = 0;
            for (int j = 0; j < N; ++j) {
                float kj = nkey[j];
                rank += (kj > myk) || (kj == myk && j < t);
            }
            cf[rank]    = myk;
            rowid[rank] = myr;
        }
        __syncthreads();
    }
    __syncthreads();

    // ---- final gather: out[c][i] = source(rowid[i]) ------------------------
    // Every source row is read exactly once across the whole dispatch and the
    // output is streamed once -> non-temporal 128-bit loads and stores.
    // Divisions hoisted out of the loop: with blockDim.x a multiple of
    // vecPerRow, each thread keeps a fixed column and strides over rows.
    const int vecPerRow = D >> 2;                  // float4 per row
    const int col       = t % vecPerRow;           // loop-invariant
    const int istep     = blockDim.x / vecPerRow;  // rows advanced per iter
    for (int i = t / vecPerRow; i < N; i += istep) {
        const int r = rowid[i];
        const floatx4* src =
            (r < N) ? (const floatx4*)(memory + ((long long)c * N + r) * D)
                    : (const floatx4*)(feats + (long long)(r - N) * D);
        floatx4 v = __builtin_nontemporal_load(src + col);
        __builtin_nontemporal_store(v, (floatx4*)(out + ((long long)c * N + i) * D) + col);
    }
}

// ---------------------------------------------------------------------------
// Host launcher
// ---------------------------------------------------------------------------
extern "C" void kernel_launch(void* const* d_in, const int* in_sizes, int n_in,
                              void* d_out, int out_size, void* d_ws, size_t ws_size,
                              hipStream_t stream) {
    const float* memory      = (const float*)d_in[0];   // [C,N,D]
    const float* confidences = (const float*)d_in[1];   // [C,N]
    const float* feats       = (const float*)d_in[2];   // [B,D]
    const float* targets     = (const float*)d_in[3];   // [B,C]
    const int*   bconf       = (const int*)d_in[4];     // [B]
    const int*   mask        = (const int*)d_in[5];     // [B]
    float*       out         = (float*)d_out;           // [C,N,D]

    const long long CND = in_sizes[0];
    const long long CN  = in_sizes[1];
    const int D = (int)(CND / CN);
    const int B = (int)(in_sizes[2] / D);
    const int C = (int)(in_sizes[3] / B);
    const int N = (int)(CN / C);

    int* tstep = (int*)d_ws;                        // B ints of scratch

    mb_argmax_kernel<<<B, TPB, 0, stream>>>(targets, mask, tstep, C);
    mb_class_kernel<<<C, TPB, 0, stream>>>(memory, confidences, feats, bconf,
                                           mask, tstep, out, N, D, B);
}